// SkeletonFilter_18794776887887
// MI455X (gfx1250) — compile-verified
//
#include <hip/hip_runtime.h>
#include <hip/hip_bf16.h>
#include <stdint.h>

#define H2     2048
#define W2     2048
#define WRDS   64                 // 2048 / 32 packed words per row
#define PADR   4                  // zero pad rows top & bottom (2 fused iterations -> halo 4)
#define PROWS  (H2 + 2 * PADR)    // 2056 padded rows
#define RPB    4                  // output rows per block in thin kernel
#define NIT    32                 // fixed full Zhang-Suen iterations (idempotent at fixed point)
#define NLAUNCH (NIT / 2)         // 2 full iterations per launch

// ---- CDNA5 async global->LDS path (guarded; falls back to plain loads) ----
#if defined(__HIP_DEVICE_COMPILE__) && __has_builtin(__builtin_amdgcn_global_load_async_to_lds_b32)
#define USE_ASYNC_LDS 1
#else
#define USE_ASYNC_LDS 0
#endif

#define AS_GLOBAL_I32(p) ((__attribute__((address_space(1))) int*)(p))
#define AS_LDS_I32(p)    ((__attribute__((address_space(3))) int*)(p))

__device__ __forceinline__ void wait_async_lds() {
#if USE_ASYNC_LDS
#if __has_builtin(__builtin_amdgcn_s_wait_asynccnt)
  __builtin_amdgcn_s_wait_asynccnt(0);
#else
  asm volatile("s_wait_asynccnt 0" ::: "memory");
#endif
#endif
}

// Bit-sliced Zhang-Suen sub-step on one 32-pixel word.
// ul/u/ur = row above (left word, center, right word), ml/m/mr = this row, dl/d/dr = row below.
// Bit i of a word = column (32*w + i); east neighbor = bit i+1, west = bit i-1.
__device__ __forceinline__ unsigned thin_word(
    unsigned ul, unsigned u, unsigned ur,
    unsigned ml, unsigned m, unsigned mr,
    unsigned dl, unsigned d, unsigned dr,
    bool first)
{
  const unsigned P2 = u;                          // N
  const unsigned P3 = (u >> 1) | (ur << 31);      // NE
  const unsigned P4 = (m >> 1) | (mr << 31);      // E
  const unsigned P5 = (d >> 1) | (dr << 31);      // SE
  const unsigned P6 = d;                          // S
  const unsigned P7 = (d << 1) | (dl >> 31);      // SW
  const unsigned P8 = (m << 1) | (ml >> 31);      // W
  const unsigned P9 = (u << 1) | (ul >> 31);      // NW

  // c1: 2 <= B <= 6  <=>  (>=2 neighbors set) && (>=2 neighbors clear)
  unsigned acc1 = 0, ge2 = 0;
  unsigned accz = 0, gez = 0;
  {
    const unsigned P[8] = {P2, P3, P4, P5, P6, P7, P8, P9};
#pragma unroll
    for (int i = 0; i < 8; ++i) {
      const unsigned p = P[i];
      ge2 |= acc1 & p;  acc1 |= p;
      const unsigned z = ~p;
      gez |= accz & z;  accz |= z;
    }
  }
  const unsigned c1 = ge2 & gez;

  // c2: A == 1, A = number of 0->1 transitions in ring P2..P9,P2
  unsigned accT = 0, ge2T = 0;
  {
    const unsigned T[8] = {~P2 & P3, ~P3 & P4, ~P4 & P5, ~P5 & P6,
                           ~P6 & P7, ~P7 & P8, ~P8 & P9, ~P9 & P2};
#pragma unroll
    for (int i = 0; i < 8; ++i) {
      ge2T |= accT & T[i];  accT |= T[i];
    }
  }
  const unsigned c2 = accT & ~ge2T;

  const unsigned c3 = first ? ~(P2 & P4 & P6) : ~(P2 & P4 & P8);
  const unsigned c4 = first ? ~(P4 & P6 & P8) : ~(P2 & P6 & P8);

  return m & ~(c1 & c2 & c3 & c4);
}

// TWO full Zhang-Suen iterations (4 sub-steps) per launch on bit-packed data.
// Block computes output rows [r0, r0+4) from input rows [r0-4, r0+8) (padded indices [r0, r0+12)).
// Stages ping-pong between LDS tiles A and B:
//   A = input (12 rows) -> S1(first)->B (10 rows) -> S2(second)->A (8 rows)
//   -> S3(first)->B (6 rows) -> S4(second)->global (4 rows).
__global__ __launch_bounds__(256) void thin_iter2_kernel(
    const unsigned* __restrict__ in, unsigned* __restrict__ out)
{
  __shared__ unsigned tA[12][WRDS + 2];   // col-halo at [0] and [65] stays zero
  __shared__ unsigned tB[10][WRDS + 2];

  const int tid = threadIdx.x;
  const int r0  = blockIdx.x * RPB;       // global output row base

  // Load 12 padded rows x 64 words into LDS (CDNA5 async global->LDS).
#pragma unroll
  for (int k = tid; k < 12 * WRDS; k += 256) {
    const int i = k >> 6, c = k & 63;
    const unsigned* gp = in + (size_t)(r0 + i) * WRDS + c;
#if USE_ASYNC_LDS
    __builtin_amdgcn_global_load_async_to_lds_b32(AS_GLOBAL_I32(gp), AS_LDS_I32(&tA[i][c + 1]), 0, 0);
#else
    tA[i][c + 1] = *gp;
#endif
  }
  if (tid < 12) { tA[tid][0] = 0u; tA[tid][WRDS + 1] = 0u; }
  if (tid < 10) { tB[tid][0] = 0u; tB[tid][WRDS + 1] = 0u; }
  wait_async_lds();
  __syncthreads();

  // S1 (first=true): 10 rows, B[j] center = global row r0-3+j
#pragma unroll
  for (int k = tid; k < 10 * WRDS; k += 256) {
    const int j = k >> 6, c = k & 63;
    tB[j][c + 1] = thin_word(tA[j][c],     tA[j][c + 1],     tA[j][c + 2],
                             tA[j + 1][c], tA[j + 1][c + 1], tA[j + 1][c + 2],
                             tA[j + 2][c], tA[j + 2][c + 1], tA[j + 2][c + 2], true);
  }
  __syncthreads();

  // S2 (first=false): 8 rows, A[j] center = global row r0-2+j
#pragma unroll
  for (int k = tid; k < 8 * WRDS; k += 256) {
    const int j = k >> 6, c = k & 63;
    tA[j][c + 1] = thin_word(tB[j][c],     tB[j][c + 1],     tB[j][c + 2],
                             tB[j + 1][c], tB[j + 1][c + 1], tB[j + 1][c + 2],
                             tB[j + 2][c], tB[j + 2][c + 1], tB[j + 2][c + 2], false);
  }
  __syncthreads();

  // S3 (first=true): 6 rows, B[j] center = global row r0-1+j
#pragma unroll
  for (int k = tid; k < 6 * WRDS; k += 256) {
    const int j = k >> 6, c = k & 63;
    tB[j][c + 1] = thin_word(tA[j][c],     tA[j][c + 1],     tA[j][c + 2],
                             tA[j + 1][c], tA[j + 1][c + 1], tA[j + 1][c + 2],
                             tA[j + 2][c], tA[j + 2][c + 1], tA[j + 2][c + 2], true);
  }
  __syncthreads();

  // S4 (first=false): 4 interior rows -> global
  const int ty = tid >> 6;          // 0..3
  const int c  = tid & 63;          // 0..63
  const unsigned v = thin_word(tB[ty][c],     tB[ty][c + 1],     tB[ty][c + 2],
                               tB[ty + 1][c], tB[ty + 1][c + 1], tB[ty + 1][c + 2],
                               tB[ty + 2][c], tB[ty + 2][c + 1], tB[ty + 2][c + 2], false);
  out[(size_t)(r0 + ty + PADR) * WRDS + c] = v;
}

// Zero the PADR top + PADR bottom padded rows of both ping-pong buffers.
__global__ __launch_bounds__(256) void zero_pads_kernel(unsigned* a, unsigned* b) {
  const int t   = blockIdx.x * 256 + threadIdx.x;     // 0 .. 2*2*PADR*WRDS-1 = 0..1023
  const int buf = t >> 9;                             // 0..1  (2*PADR*WRDS = 512 per buffer)
  const int k   = t & 511;                            // 8 rows * 64 words
  const int row = k >> 6;                             // 0..7
  const int c   = k & 63;
  const int pr  = (row < PADR) ? row : (PROWS - 2 * PADR + row);
  unsigned* p = buf ? b : a;
  p[pr * WRDS + c] = 0u;
}

// Grayscale + threshold + bit-pack (one wave32 ballot -> one packed word).
__global__ __launch_bounds__(256) void binarize_pack_kernel(
    const float* __restrict__ img, unsigned* __restrict__ out)
{
  const int idx = blockIdx.x * 256 + threadIdx.x;     // pixel index 0..4M-1
  const float r  = img[idx];
  const float g  = img[idx + H2 * W2];
  const float bl = img[idx + 2 * H2 * W2];
  const float gray = 0.2989f * r + 0.587f * g + 0.114f * bl;
  const bool on = gray > 0.5f;
  unsigned m;
#if defined(__HIP_DEVICE_COMPILE__) && __has_builtin(__builtin_amdgcn_ballot_w32)
  m = __builtin_amdgcn_ballot_w32(on);
#else
  m = (unsigned)__ballot(on);
#endif
  if ((threadIdx.x & 31) == 0) {
    out[(idx >> 5) + PADR * WRDS] = m;                // padded row offset
  }
}

// Unpack bits -> int32 {0,1}.
__global__ __launch_bounds__(256) void unpack_kernel(
    const unsigned* __restrict__ in, int* __restrict__ out)
{
  const int idx = blockIdx.x * 256 + threadIdx.x;     // 0..4M-1
  const int r = idx >> 11;
  const int c = idx & 2047;
  const unsigned w = in[(size_t)(r + PADR) * WRDS + (c >> 5)];
  out[idx] = (int)((w >> (c & 31)) & 1u);
}

extern "C" void kernel_launch(void* const* d_in, const int* in_sizes, int n_in,
                              void* d_out, int out_size, void* d_ws, size_t ws_size,
                              hipStream_t stream) {
  (void)in_sizes; (void)n_in; (void)out_size; (void)ws_size;
  const float* img = (const float*)d_in[0];
  int* out = (int*)d_out;

  unsigned* bufA = (unsigned*)d_ws;
  unsigned* bufB = bufA + (size_t)PROWS * WRDS;

  zero_pads_kernel<<<4, 256, 0, stream>>>(bufA, bufB);
  binarize_pack_kernel<<<(H2 * W2) / 256, 256, 0, stream>>>(img, bufA);

  unsigned* cur = bufA;
  unsigned* nxt = bufB;
  for (int i = 0; i < NLAUNCH; ++i) {
    thin_iter2_kernel<<<H2 / RPB, 256, 0, stream>>>(cur, nxt);
    unsigned* t = cur; cur = nxt; nxt = t;
  }

  unpack_kernel<<<(H2 * W2) / 256, 256, 0, stream>>>(cur, out);
}